// GridGatingUnit_25254407700482
// MI455X (gfx1250) — compile-verified
//
#include <hip/hip_runtime.h>

// GridGatingUnit for MI455X (gfx1250, wave32).
// Memory-bound (~384-512 MB moved vs 4.3 GFLOP -> HBM roofline ~17-22 us):
//   Pass 1: channel-LayerNorm statistics per (b,h,w) -> d_ws (v gets L2-resident).
//   Pass 2: per (b,c) 64x64x64 GEMM via V_WMMA_F32_16X16X4_F32, layernorm
//           affine fused on load, u*(v+1) gate fused on a fully-coalesced
//           LDS-staged epilogue. lin_w staged via async global->LDS;
//           u tile prefetched behind the WMMA loop.

typedef float v2f __attribute__((ext_vector_type(2)));
typedef float v8f __attribute__((ext_vector_type(8)));
typedef int   v4i __attribute__((vector_size(16)));   // matches builtin's pointee

#define BATCH 32
#define CTWO 512
#define CIN 256
#define HS 64
#define HW (HS * HS)       // 4096
#define LDS_STRIDE 65      // 64 + 1 pad: conflict-free A & B fragment reads

// ---------------------------------------------------------------------------
// Pass 1: mean / rstd over the channel dim for each (b,h,w) point.
// One thread per point; channel loop is lane-coalesced (32 lanes = 128B).
// ---------------------------------------------------------------------------
__global__ __launch_bounds__(256) void ggu_stats(const float* __restrict__ x,
                                                 float* __restrict__ mean_out,
                                                 float* __restrict__ rstd_out) {
    const int p  = blockIdx.x * 256 + threadIdx.x;   // [0, B*HW)
    const int b  = p >> 12;                          // / 4096
    const int hw = p & (HW - 1);
    const float* base = x + ((size_t)(b * CTWO + CIN)) * HW + hw;   // v half

    float s = 0.f, ss = 0.f;
#pragma unroll 8
    for (int c = 0; c < CIN; ++c) {
        float v = base[(size_t)c * HW];
        s  += v;
        ss += v * v;
    }
    const float mean = s * (1.0f / CIN);
    const float var  = ss * (1.0f / CIN) - mean * mean;
    mean_out[p] = mean;
    rstd_out[p] = rsqrtf(var + 1e-6f);
}

// ---------------------------------------------------------------------------
// Pass 2: per (b,c): v_norm = LN(v); D = lin_w @ v_norm (64x64x64 GEMM);
//         out = u * (D + lin_b + 1).
// 256 threads = 8 waves; each wave owns two 16x16 output tiles and runs the
// K=64 loop in steps of 4 with v_wmma_f32_16x16x4_f32 (f32-exact).
// ---------------------------------------------------------------------------
__global__ __launch_bounds__(256) void ggu_gemm(const float* __restrict__ x,
                                                const float* __restrict__ norm_w,
                                                const float* __restrict__ norm_b,
                                                const float* __restrict__ lin_w,
                                                const float* __restrict__ lin_b,
                                                const float* __restrict__ mean_in,
                                                const float* __restrict__ rstd_in,
                                                float* __restrict__ out) {
    __shared__ float vs[HS * LDS_STRIDE];   // normalized v tile [h][w]; later out tile
    __shared__ float lw[HS * LDS_STRIDE];   // lin_w [k][h]

    const int tid = threadIdx.x;
    const int bc  = blockIdx.x;
    const int b   = bc >> 8;                 // / CIN
    const int c   = bc & (CIN - 1);

    const float gscale = norm_w[c];
    const float gbias  = norm_b[c];

    const float* vsrc = x + ((size_t)(b * CTWO + CIN + c)) * HW;
    const float* usrc = x + ((size_t)(b * CTWO + c)) * HW;
    const float* mb   = mean_in + (size_t)b * HW;
    const float* rb   = rstd_in + (size_t)b * HW;

    // Prefetch the u gate tile (16 KB) so the epilogue's reads hit cache;
    // lowers to global_prefetch_b8 (one byte per 64B slice covers all lines).
    __builtin_prefetch(usrc + tid * 16, 0, 1);

    // Stage lin_w (16 KB) into LDS via the gfx1250 async global->LDS path.
#if __has_builtin(__builtin_amdgcn_global_load_async_to_lds_b128)
    {
        typedef __attribute__((address_space(1))) v4i* g4_t;
        typedef __attribute__((address_space(3))) v4i* l4_t;
#pragma unroll
        for (int j = 0; j < 4; ++j) {
            const int i16 = (tid + j * 256) * 4;         // float index, 16B chunks
            const int h = i16 >> 6, w = i16 & 63;        // w%4==0, never crosses rows
            __builtin_amdgcn_global_load_async_to_lds_b128(
                (g4_t)(unsigned long long)(lin_w + i16),
                (l4_t)(unsigned int)(unsigned long long)&lw[h * LDS_STRIDE + w],
                0, 0);
        }
    }
#else
#pragma unroll
    for (int j = 0; j < 4; ++j) {
        const int i16 = (tid + j * 256) * 4;
        const int h = i16 >> 6, w = i16 & 63;
        const float4 t = *(const float4*)(lin_w + i16);
        lw[h * LDS_STRIDE + w + 0] = t.x;
        lw[h * LDS_STRIDE + w + 1] = t.y;
        lw[h * LDS_STRIDE + w + 2] = t.z;
        lw[h * LDS_STRIDE + w + 3] = t.w;
    }
#endif

    // Load + normalize the v tile into LDS (coalesced 128B per wave).
#pragma unroll
    for (int j = 0; j < 16; ++j) {
        const int i = tid + j * 256;         // [0, 4096)
        const int h = i >> 6, w = i & 63;
        const float v = vsrc[i];
        vs[h * LDS_STRIDE + w] = (v - mb[i]) * rb[i] * gscale + gbias;
    }

#if __has_builtin(__builtin_amdgcn_global_load_async_to_lds_b128)
#if __has_builtin(__builtin_amdgcn_s_wait_asynccnt)
    __builtin_amdgcn_s_wait_asynccnt(0);
#else
    asm volatile("s_wait_asynccnt 0" ::: "memory");
#endif
#endif
    __syncthreads();

    const int lane = tid & 31;
    const int wid  = tid >> 5;               // 0..7
    const int tm   = (wid & 3) * 16;         // output row-tile base (k dim)
    const int tn   = (wid >> 2) * 32;        // output col base: 2 tiles wide

    const int lrow  = lane & 15;
    const int khalf = (lane >> 4) * 2;       // lanes 0-15 -> K+{0,1}; 16-31 -> K+{2,3}

    v8f acc0 = {};
    v8f acc1 = {};

#pragma unroll 4
    for (int k0 = 0; k0 < HS; k0 += 4) {
        // A fragment (16x4 of lin_w): lane holds row tm+lrow, K = k0+khalf+{0,1}
        v2f a;
        a.x = lw[(tm + lrow) * LDS_STRIDE + k0 + khalf + 0];
        a.y = lw[(tm + lrow) * LDS_STRIDE + k0 + khalf + 1];
        // B fragments (4x16 of v_norm): rows striped across lanes per VGPR
        v2f b0, b1;
        b0.x = vs[(k0 + khalf + 0) * LDS_STRIDE + tn + lrow];
        b0.y = vs[(k0 + khalf + 1) * LDS_STRIDE + tn + lrow];
        b1.x = vs[(k0 + khalf + 0) * LDS_STRIDE + tn + 16 + lrow];
        b1.y = vs[(k0 + khalf + 1) * LDS_STRIDE + tn + 16 + lrow];

        acc0 = __builtin_amdgcn_wmma_f32_16x16x4_f32(
            false, a, false, b0, (short)0, acc0, false, false);
        acc1 = __builtin_amdgcn_wmma_f32_16x16x4_f32(
            false, a, false, b1, (short)0, acc1, false, false);
    }

    // Stage D + lin_b + 1 back into the (now dead) v tile so the gate/store
    // phase is fully coalesced.
    __syncthreads();
    {
        const int nofs  = lane & 15;
        const int mhalf = (lane >> 4) * 8;
#pragma unroll
        for (int j = 0; j < 8; ++j) {
            const int m = tm + mhalf + j;
            const float bias = lin_b[m] + 1.0f;
            vs[m * LDS_STRIDE + tn + nofs]      = acc0[j] + bias;
            vs[m * LDS_STRIDE + tn + 16 + nofs] = acc1[j] + bias;
        }
    }
    __syncthreads();

    // Coalesced epilogue: 512B contiguous per wave per step (float4/lane).
    float* dst = out + ((size_t)(b * CIN + c)) * HW;
#pragma unroll
    for (int j = 0; j < 4; ++j) {
        const int i4 = (tid + j * 256) * 4;              // float index
        const int h = i4 >> 6, w = i4 & 63;              // w%4==0
        const float4 u4 = *(const float4*)(usrc + i4);
        float4 o;
        o.x = u4.x * vs[h * LDS_STRIDE + w + 0];
        o.y = u4.y * vs[h * LDS_STRIDE + w + 1];
        o.z = u4.z * vs[h * LDS_STRIDE + w + 2];
        o.w = u4.w * vs[h * LDS_STRIDE + w + 3];
        *(float4*)(dst + i4) = o;
    }
}

// ---------------------------------------------------------------------------
extern "C" void kernel_launch(void* const* d_in, const int* in_sizes, int n_in,
                              void* d_out, int out_size, void* d_ws, size_t ws_size,
                              hipStream_t stream) {
    const float* x      = (const float*)d_in[0];
    const float* norm_w = (const float*)d_in[1];
    const float* norm_b = (const float*)d_in[2];
    const float* lin_w  = (const float*)d_in[3];
    const float* lin_b  = (const float*)d_in[4];

    float* mean = (float*)d_ws;                      // B*HW floats
    float* rstd = mean + (size_t)BATCH * HW;         // B*HW floats (1 MB total)

    ggu_stats<<<(BATCH * HW) / 256, 256, 0, stream>>>(x, mean, rstd);
    ggu_gemm<<<BATCH * CIN, 256, 0, stream>>>(x, norm_w, norm_b, lin_w, lin_b,
                                              mean, rstd, (float*)d_out);
}